// PromptDualAttentionFreq_3143916061229
// MI455X (gfx1250) — compile-verified
//
#include <hip/hip_runtime.h>
#include <math.h>

#define C2v 512
#define Tv  512
#define Bv  32
#define Ev  256
#define Mv  8
#define Nv  5
#define Rv  64
#define MNv 40
#define BTv (Bv*Tv)     // 16384
#define TCv (Tv*C2v)    // 262144

typedef __attribute__((ext_vector_type(2))) float v2f;
typedef __attribute__((ext_vector_type(8))) float v8f;

// ---------------------------------------------------------------------------
// 1) reshape (B,E,T,2)->(B,T,C2) + LayerNorm over C2
// ---------------------------------------------------------------------------
__global__ void __launch_bounds__(256) ln_kernel(const float* __restrict__ x_ri,
                                                 const float* __restrict__ g,
                                                 const float* __restrict__ beta,
                                                 float* __restrict__ xn) {
  __shared__ float tile[16][C2v + 4];
  __shared__ float redS[256], redQ[256];
  __shared__ float mu[16], rs[16];
  int b = blockIdx.y, t0 = blockIdx.x * 16;
  int tid = threadIdx.x;
  int eo = tid >> 4, tt = tid & 15;
  for (int e0 = 0; e0 < Ev; e0 += 16) {
    int e = e0 + eo;
    const float* p = x_ri + (((size_t)(b * Ev + e)) * Tv + (t0 + tt)) * 2;
    float2 v = *(const float2*)p;
    tile[tt][2 * e]     = v.x;
    tile[tt][2 * e + 1] = v.y;
  }
  __syncthreads();
  {
    int row = tid >> 4, l = tid & 15;
    float s = 0.f, q = 0.f;
    for (int i = 0; i < 32; i++) { float v = tile[row][l * 32 + i]; s += v; q += v * v; }
    redS[tid] = s; redQ[tid] = q;
  }
  __syncthreads();
  if (tid < 16) {
    float s = 0.f, q = 0.f;
    for (int i = 0; i < 16; i++) { s += redS[tid * 16 + i]; q += redQ[tid * 16 + i]; }
    float m = s * (1.f / 512.f);
    float var = q * (1.f / 512.f) - m * m;
    mu[tid] = m;
    rs[tid] = rsqrtf(var + 1e-5f);
  }
  __syncthreads();
  for (int idx = tid; idx < 16 * C2v; idx += 256) {
    int r = idx >> 9, c = idx & 511;
    float v = (tile[r][c] - mu[r]) * rs[r] * g[c] + beta[c];
    xn[(((size_t)b) * Tv + (t0 + r)) * C2v + c] = v;
  }
}

// ---------------------------------------------------------------------------
// 2) column sums over T:  S[seq,c] = sum_t X[seq,t,c]
// ---------------------------------------------------------------------------
__global__ void colsum_kernel(const float* __restrict__ X, float* __restrict__ S, int nseq) {
  int id = blockIdx.x * 256 + threadIdx.x;
  if (id >= nseq * C2v) return;
  int seq = id >> 9, c = id & 511;
  const float* p = X + (size_t)seq * TCv + c;
  float s = 0.f;
  for (int t = 0; t < Tv; t++) s += p[(size_t)t * C2v];
  S[id] = s;
}

// ---------------------------------------------------------------------------
// 3) generic row GEMM: out[r,j] = A[r,:] @ W[:,j] + bias[j]*biasScale
// ---------------------------------------------------------------------------
__global__ void rowgemm_kernel(const float* __restrict__ A, long aStride, int rows,
                               const float* __restrict__ W, const float* __restrict__ bias,
                               float biasScale, float* __restrict__ out) {
  int id = blockIdx.x * 256 + threadIdx.x;
  if (id >= rows * C2v) return;
  int r = id >> 9, j = id & 511;
  const float* a = A + (size_t)r * aStride;
  float acc = bias ? bias[j] * biasScale : 0.f;
  for (int i = 0; i < C2v; i++) acc = fmaf(a[i], W[(size_t)i * C2v + j], acc);
  out[id] = acc;
}

// 512x512 transpose (LDS tiled): out[j,i] = in[i,j]
__global__ void __launch_bounds__(256) transpose_kernel(const float* __restrict__ in,
                                                        float* __restrict__ out) {
  __shared__ float t[32][33];
  int bx = blockIdx.x * 32, by = blockIdx.y * 32;
  int x = threadIdx.x & 31, y0 = threadIdx.x >> 5;   // 32 x 8
  for (int yy = y0; yy < 32; yy += 8)
    t[yy][x] = in[(size_t)(by + yy) * C2v + bx + x];
  __syncthreads();
  for (int yy = y0; yy < 32; yy += 8)
    out[(size_t)(bx + yy) * C2v + by + x] = t[x][yy];
}

// ---------------------------------------------------------------------------
// 4) pool stage 1: rr[seq,r] = (1/T)*sum_c[(w0+w1+w2)S - w0*xL - w2*x0] + conv_b[r]
// ---------------------------------------------------------------------------
__global__ void poolr_kernel(const float* __restrict__ S,
                             const float* __restrict__ X0, long x0Stride,
                             const float* __restrict__ XL, long xLStride,
                             const float* __restrict__ conv_w, const float* __restrict__ conv_b,
                             float* __restrict__ rr, int nseq) {
  int id = blockIdx.x * 64 + threadIdx.x;
  if (id >= nseq * Rv) return;
  int seq = id >> 6, r = id & 63;
  const float* s  = S  + (size_t)seq * C2v;
  const float* x0 = X0 + (size_t)seq * x0Stride;
  const float* xl = XL + (size_t)seq * xLStride;
  const float* w  = conv_w + (size_t)r * C2v * 3;
  float acc = 0.f;
  for (int c = 0; c < C2v; c++) {
    float w0 = w[3 * c], w1 = w[3 * c + 1], w2 = w[3 * c + 2];
    acc += (w0 + w1 + w2) * s[c] - w0 * xl[c] - w2 * x0[c];
  }
  rr[id] = acc * (1.f / 512.f) + conv_b[r];
}

// pool stage 2: out[seq,j] = rr[seq,:] @ pool_w[:,j] + pool_b[j]
__global__ void poolfinish_kernel(const float* __restrict__ rr,
                                  const float* __restrict__ pool_w,
                                  const float* __restrict__ pool_b,
                                  float* __restrict__ out, int nseq) {
  int id = blockIdx.x * 256 + threadIdx.x;
  if (id >= nseq * C2v) return;
  int seq = id >> 9, j = id & 511;
  float acc = pool_b[j];
  for (int r = 0; r < Rv; r++) acc = fmaf(rr[seq * Rv + r], pool_w[(size_t)r * C2v + j], acc);
  out[id] = acc;
}

// ---------------------------------------------------------------------------
// 5) inner attention: sc[b,m,n]=q_prot.keys_p, masked softmax over n
// ---------------------------------------------------------------------------
__global__ void aw1_kernel(const float* __restrict__ qprot, const float* __restrict__ keysp,
                           const unsigned char* __restrict__ mask, float* __restrict__ aw1) {
  int bm = blockIdx.x;            // b*8+m
  int b = bm >> 3, m = bm & 7;
  int lane = threadIdx.x;
  const float* q = qprot + (size_t)b * C2v;
  float p[Nv];
  for (int n = 0; n < Nv; n++) {
    const float* kk = keysp + ((size_t)m * Nv + n) * C2v;
    float s = 0.f;
    for (int c = lane; c < C2v; c += 32) s += q[c] * kk[c];
    for (int off = 16; off; off >>= 1) s += __shfl_down(s, off, 32);
    p[n] = s;
  }
  if (lane == 0) {
    const float scale = 0.04419417382415922f;  // 1/sqrt(512)
    float v[Nv]; unsigned char mk[Nv];
    float mx = -INFINITY;
    for (int n = 0; n < Nv; n++) {
      mk[n] = mask[m * Nv + n];
      v[n] = mk[n] ? p[n] * scale : -INFINITY;
      if (v[n] > mx) mx = v[n];
    }
    float sum = 0.f;
    for (int n = 0; n < Nv; n++) { v[n] = mk[n] ? __expf(v[n] - mx) : 0.f; sum += v[n]; }
    for (int n = 0; n < Nv; n++) aw1[(size_t)bm * Nv + n] = v[n] / sum;
  }
}

// y_ks[b,m,c] = sum_n aw1[b,m,n] * pooled_vals[m,n,c]
__global__ void yks_kernel(const float* __restrict__ aw1, const float* __restrict__ pv,
                           float* __restrict__ yks) {
  int id = blockIdx.x * 256 + threadIdx.x;   // Bv*Mv*C2v
  int c = id & 511;
  int bm = id >> 9;
  int m = bm & 7;
  float s = 0.f;
  for (int n = 0; n < Nv; n++) s = fmaf(aw1[bm * Nv + n], pv[((size_t)m * Nv + n) * C2v + c], s);
  yks[id] = s;
}

// ---------------------------------------------------------------------------
// 6) outer attention softmax + combined weights w[b,m,n]=aw2*aw1
// ---------------------------------------------------------------------------
__global__ void __launch_bounds__(256) aw2_kernel(const float* __restrict__ qv,
                                                  const float* __restrict__ kv,
                                                  const float* __restrict__ aw1,
                                                  float* __restrict__ aw2_out,
                                                  float* __restrict__ wcomb) {
  int b = blockIdx.x;
  int tid = threadIdx.x;
  int wid = tid >> 5, lane = tid & 31;
  __shared__ float sc2[Mv], a2[Mv];
  const float* q = qv + (size_t)b * C2v;
  const float* k = kv + ((size_t)b * Mv + wid) * C2v;
  float s = 0.f;
  for (int c = lane; c < C2v; c += 32) s += q[c] * k[c];
  for (int off = 16; off; off >>= 1) s += __shfl_down(s, off, 32);
  if (lane == 0) sc2[wid] = s * 0.04419417382415922f;
  __syncthreads();
  if (tid == 0) {
    float mx = sc2[0];
    for (int m = 1; m < Mv; m++) mx = fmaxf(mx, sc2[m]);
    float e[Mv]; float sum = 0.f;
    for (int m = 0; m < Mv; m++) { e[m] = __expf(sc2[m] - mx); sum += e[m]; }
    for (int m = 0; m < Mv; m++) { a2[m] = e[m] / sum; aw2_out[b * Mv + m] = a2[m]; }
  }
  __syncthreads();
  if (tid < MNv) wcomb[b * MNv + tid] = a2[tid / Nv] * aw1[(size_t)b * MNv + tid];
}

// ---------------------------------------------------------------------------
// 7) P_bar[b,t,c] = sum_{mn} w[b,mn] * protos[mn,t,c]  (protos read exactly once)
// ---------------------------------------------------------------------------
__global__ void __launch_bounds__(256) pbar_kernel(const float* __restrict__ protos,
                                                   const float* __restrict__ wcomb,
                                                   float* __restrict__ Pbar) {
  __shared__ float ws[Bv * MNv];
  int tid = threadIdx.x;
  for (int i = tid; i < Bv * MNv; i += 256) ws[i] = wcomb[i];
  __syncthreads();
  size_t tc = (size_t)blockIdx.x * 256 + tid;     // 0 .. T*C2-1
  float acc[Bv];
#pragma unroll
  for (int b = 0; b < Bv; b++) acc[b] = 0.f;
  for (int mn = 0; mn < MNv; mn++) {
    float p = protos[(size_t)mn * TCv + tc];
#pragma unroll
    for (int b = 0; b < Bv; b++) acc[b] = fmaf(ws[b * MNv + mn], p, acc[b]);
  }
  for (int b = 0; b < Bv; b++) Pbar[(size_t)b * TCv + tc] = acc[b];
}

// ---------------------------------------------------------------------------
// 8) fused final WMMA GEMM with TRANSPOSED weights:
//    out = scatter( Pbar@WWt^T + xn@wyT^T + bfin )
//    Each wave owns a 32x64 tile: 8 accumulators; every B fragment (1 b64 load)
//    feeds 2 WMMAs, every A fragment feeds 4.
// ---------------------------------------------------------------------------
__global__ void __launch_bounds__(256) final_gemm_kernel(
    const float* __restrict__ Pbar, const float* __restrict__ WWt,
    const float* __restrict__ xn,   const float* __restrict__ wyT,
    const float* __restrict__ bfin, float* __restrict__ out) {
  int wave = threadIdx.x >> 5;
  int lane = threadIdx.x & 31;
  int half = lane >> 4, ln = lane & 15;
  long row0 = (long)blockIdx.x * 256 + wave * 32;   // 16384/256 = 64 blocks in x
  int col0 = blockIdx.y * 64;                        // 512/64 = 8 blocks in y

  v8f acc[8];
#pragma unroll
  for (int j = 0; j < 4; j++) {
    float bvv = bfin[col0 + 16 * j + ln];
#pragma unroll
    for (int r = 0; r < 8; r++) { acc[j][r] = bvv; acc[4 + j][r] = bvv; }
  }

  for (int pass = 0; pass < 2; pass++) {
    const float* A  = pass ? xn : Pbar;
    const float* Wt = pass ? wyT : WWt;      // Wt[col*512 + k] = W[k*512 + col]
    const float* ap0 = A + (row0 + ln) * 512;
    const float* ap1 = A + (row0 + 16 + ln) * 512;
    for (int k0 = 0; k0 < 512; k0 += 4) {
      int kk = k0 + 2 * half;
      v2f a0 = *(const v2f*)(ap0 + kk);
      v2f a1 = *(const v2f*)(ap1 + kk);
#pragma unroll
      for (int j = 0; j < 4; j++) {
        v2f bf = *(const v2f*)(Wt + (size_t)(col0 + 16 * j + ln) * 512 + kk);
        acc[j]     = __builtin_amdgcn_wmma_f32_16x16x4_f32(
            false, a0, false, bf, (short)0, acc[j], false, false);
        acc[4 + j] = __builtin_amdgcn_wmma_f32_16x16x4_f32(
            false, a1, false, bf, (short)0, acc[4 + j], false, false);
      }
    }
  }

  // scatter store: (b,t,c) -> out[b,e,t,ri]  with e=c>>1, ri=c&1
#pragma unroll
  for (int j = 0; j < 4; j++) {
    int c = col0 + 16 * j + ln;
    int e = c >> 1, ri = c & 1;
#pragma unroll
    for (int rt = 0; rt < 2; rt++) {
#pragma unroll
      for (int r = 0; r < 8; r++) {
        long grow = row0 + 16 * rt + r + 8 * half;
        long b = grow >> 9, t = grow & 511;
        out[(((size_t)b * Ev + e) * Tv + t) * 2 + ri] = acc[4 * rt + j][r];
      }
    }
  }
}

// ---------------------------------------------------------------------------
extern "C" void kernel_launch(void* const* d_in, const int* in_sizes, int n_in,
                              void* d_out, int out_size, void* d_ws, size_t ws_size,
                              hipStream_t stream) {
  const float* x_ri   = (const float*)d_in[0];
  const float* protos = (const float*)d_in[1];
  const unsigned char* pmask = (const unsigned char*)d_in[2];
  const float* conv_w = (const float*)d_in[3];
  const float* conv_b = (const float*)d_in[4];
  const float* pool_w = (const float*)d_in[5];
  const float* pool_b = (const float*)d_in[6];
  const float* ln_g   = (const float*)d_in[7];
  const float* ln_b   = (const float*)d_in[8];
  const float* wqp = (const float*)d_in[9];   const float* bqp = (const float*)d_in[10];
  const float* wkp = (const float*)d_in[11];  const float* bkp = (const float*)d_in[12];
  const float* wvp = (const float*)d_in[13];  const float* bvp = (const float*)d_in[14];
  const float* wq  = (const float*)d_in[15];  const float* bq  = (const float*)d_in[16];
  const float* wk  = (const float*)d_in[17];  const float* bk  = (const float*)d_in[18];
  const float* wv  = (const float*)d_in[19];  const float* bv  = (const float*)d_in[20];
  const float* wy  = (const float*)d_in[21];  const float* by  = (const float*)d_in[22];
  float* out = (float*)d_out;

  float* w = (float*)d_ws;
  float* xn    = w; w += (size_t)BTv * C2v;
  float* Pbar  = w; w += (size_t)BTv * C2v;
  float* Sb    = w; w += Bv * C2v;
  float* Smn   = w; w += MNv * C2v;
  float* Sv    = w; w += MNv * C2v;
  float* v0    = w; w += MNv * C2v;
  float* vL    = w; w += MNv * C2v;
  float* rr    = w; w += MNv * Rv;
  float* xq    = w; w += Bv * C2v;
  float* pks   = w; w += MNv * C2v;
  float* pv    = w; w += MNv * C2v;
  float* qprot = w; w += Bv * C2v;
  float* keysp = w; w += MNv * C2v;
  float* aw1   = w; w += Bv * MNv;
  float* yks   = w; w += Bv * Mv * C2v;
  float* qv    = w; w += Bv * C2v;
  float* kv    = w; w += Bv * Mv * C2v;
  float* wcomb = w; w += Bv * MNv;
  float* Wtmp  = w; w += C2v * C2v;   // wvp@wv, later reused as WWt (transposed WW)
  float* WWm   = w; w += C2v * C2v;   // WW row-major
  float* wyT   = w; w += C2v * C2v;   // wy transposed
  float* btmp  = w; w += C2v;
  float* bfin  = w; w += C2v;

  // 1) layernorm
  ln_kernel<<<dim3(Tv / 16, Bv), 256, 0, stream>>>(x_ri, ln_g, ln_b, xn);

  // weight folding: WW = wvp@wv@wy ; bfin = (bvp@wv+bv)@wy + by ; transposes
  rowgemm_kernel<<<(C2v * C2v) / 256, 256, 0, stream>>>(wvp, C2v, C2v, wv, (const float*)nullptr, 0.f, Wtmp);
  rowgemm_kernel<<<(C2v * C2v) / 256, 256, 0, stream>>>(Wtmp, C2v, C2v, wy, (const float*)nullptr, 0.f, WWm);
  transpose_kernel<<<dim3(16, 16), 256, 0, stream>>>(WWm, Wtmp);   // Wtmp := WW^T
  transpose_kernel<<<dim3(16, 16), 256, 0, stream>>>(wy, wyT);     // wyT  := wy^T
  rowgemm_kernel<<<2, 256, 0, stream>>>(bvp, 0, 1, wv, bv, 1.f, btmp);
  rowgemm_kernel<<<2, 256, 0, stream>>>(btmp, 0, 1, wy, by, 1.f, bfin);

  // 2) column sums
  colsum_kernel<<<(Bv * C2v) / 256, 256, 0, stream>>>(xn, Sb, Bv);
  colsum_kernel<<<(MNv * C2v) / 256, 256, 0, stream>>>(protos, Smn, MNv);

  // 3) x_q = pool(xn)
  poolr_kernel<<<Bv, 64, 0, stream>>>(Sb, xn, (long)TCv, xn + (size_t)(Tv - 1) * C2v, (long)TCv,
                                      conv_w, conv_b, rr, Bv);
  poolfinish_kernel<<<(Bv * C2v) / 256, 256, 0, stream>>>(rr, pool_w, pool_b, xq, Bv);

  // 4) prompts_ks = pool(protos)
  poolr_kernel<<<MNv, 64, 0, stream>>>(Smn, protos, (long)TCv, protos + (size_t)(Tv - 1) * C2v, (long)TCv,
                                       conv_w, conv_b, rr, MNv);
  poolfinish_kernel<<<(MNv * C2v) / 256, 256, 0, stream>>>(rr, pool_w, pool_b, pks, MNv);

  // 5) pooled vals pv = pool(protos@wvp + bvp)  via affine sums/first/last rows
  rowgemm_kernel<<<(MNv * C2v) / 256, 256, 0, stream>>>(Smn, C2v, MNv, wvp, bvp, 512.f, Sv);
  rowgemm_kernel<<<(MNv * C2v) / 256, 256, 0, stream>>>(protos, TCv, MNv, wvp, bvp, 1.f, v0);
  rowgemm_kernel<<<(MNv * C2v) / 256, 256, 0, stream>>>(protos + (size_t)(Tv - 1) * C2v, TCv, MNv, wvp, bvp, 1.f, vL);
  poolr_kernel<<<MNv, 64, 0, stream>>>(Sv, v0, (long)C2v, vL, (long)C2v, conv_w, conv_b, rr, MNv);
  poolfinish_kernel<<<(MNv * C2v) / 256, 256, 0, stream>>>(rr, pool_w, pool_b, pv, MNv);

  // 6) inner attention
  rowgemm_kernel<<<(Bv * C2v) / 256, 256, 0, stream>>>(xq, C2v, Bv, wqp, bqp, 1.f, qprot);
  rowgemm_kernel<<<(MNv * C2v) / 256, 256, 0, stream>>>(pks, C2v, MNv, wkp, bkp, 1.f, keysp);
  aw1_kernel<<<Bv * Mv, 32, 0, stream>>>(qprot, keysp, pmask, aw1);
  yks_kernel<<<(Bv * Mv * C2v) / 256, 256, 0, stream>>>(aw1, pv, yks);

  // 7) outer attention
  rowgemm_kernel<<<(Bv * C2v) / 256, 256, 0, stream>>>(xq, C2v, Bv, wq, bq, 1.f, qv);
  rowgemm_kernel<<<(Bv * Mv * C2v) / 256, 256, 0, stream>>>(yks, C2v, Bv * Mv, wk, bk, 1.f, kv);
  aw2_kernel<<<Bv, 256, 0, stream>>>(qv, kv, aw1, out + (size_t)Bv * Ev * Tv * 2, wcomb);

  // 8) proto combine + fused final WMMA GEMMs with scatter to (B,E,T,2)
  pbar_kernel<<<TCv / 256, 256, 0, stream>>>(protos, wcomb, Pbar);
  final_gemm_kernel<<<dim3(BTv / 256, C2v / 64), 256, 0, stream>>>(Pbar, Wtmp, xn, wyT, bfin, out);
}